// Window_77790447665189
// MI455X (gfx1250) — compile-verified
//
#include <hip/hip_runtime.h>
#include <hip/hip_bf16.h>

typedef __attribute__((ext_vector_type(2))) float v2f;
typedef __attribute__((ext_vector_type(8))) float v8f;

#define B_SZ    4096
#define L_SZ    256
#define V_SZ    80
#define CELL_SZ 512
#define K_SZ    10
#define N3K     30   // 3*K
#define NPAD    32   // padded N

// ---------------------------------------------------------------------------
// Kernel 1: params = exp(x @ W + b); kappa = kappa_old + pre_kappa.
// fp32 WMMA (V_WMMA_F32_16X16X4_F32). W (512x30) is staged once per block
// into LDS, zero-padded to 32 columns AND k-pair interleaved:
//   Wp[(k>>1)*64 + 2*n + (k&1)]
// so each B fragment {W[k][n], W[k+1][n]} is one aligned ds_load_b64 landing
// directly in an even VGPR pair (no repack movs before v_wmma).
// ws_params per row (32 floats): [0..9]=alpha [10..19]=beta [20..29]=kappa
// ---------------------------------------------------------------------------
__global__ __launch_bounds__(256) void window_gemm_wmma(
    const float* __restrict__ x, const float* __restrict__ W,
    const float* __restrict__ bias, const float* __restrict__ kappa_old,
    float* __restrict__ ws_params, float* __restrict__ out_kappa)
{
    __shared__ float Wp[CELL_SZ * NPAD];   // 64 KB, pair-interleaved

    const int t = threadIdx.x;

    // Stage W into LDS: dest = (k>>1)*64 + 2n + (k&1); cols 30,31 = 0.
    for (int idx = t; idx < CELL_SZ * NPAD; idx += 256) {
        const int k = idx >> 5;
        const int n = idx & 31;
        const int dst = ((k >> 1) << 6) + (n << 1) + (k & 1);
        Wp[dst] = (n < N3K) ? W[k * N3K + n] : 0.0f;
    }
    __syncthreads();

    const int lane = t & 31;
    const int wave = t >> 5;
    const int row0 = (blockIdx.x * 8 + wave) * 16;

    // A-fragment (16x4 f32): lanes 0-15 -> M=lane, K pair {0,1};
    // lanes 16-31 -> M=lane-16, K pair {2,3}.
    const int m  = lane & 15;
    const int kq = (lane >> 4) * 2;
    // B-fragment (4x16 f32): lanes 0-15 carry K rows {0,1} for N=lane,
    // lanes 16-31 carry K rows {2,3}.
    const int n0 = lane & 15;

    // Per-lane LDS base: pair (kq>>1) plus column 2*n0 (tile 1 at +32).
    const float* wl = Wp + ((lane >> 4) << 6) + (n0 << 1);

    v8f acc0 = {};
    v8f acc1 = {};

    const float* xrow = x + (size_t)(row0 + m) * CELL_SZ;

    for (int k0 = 0; k0 < CELL_SZ; k0 += 4) {
        v2f a;
        a.x = xrow[k0 + kq];
        a.y = xrow[k0 + kq + 1];

        const int pbase = (k0 >> 1) << 6;      // (k0/2)*64 floats
        const v2f b0 = *(const v2f*)(wl + pbase);        // {W[k][n0],  W[k+1][n0]}
        const v2f b1 = *(const v2f*)(wl + pbase + 32);   // {W[k][n0+16],W[k+1][n0+16]}

        acc0 = __builtin_amdgcn_wmma_f32_16x16x4_f32(
            false, a, false, b0, (short)0, acc0, false, false);
        acc1 = __builtin_amdgcn_wmma_f32_16x16x4_f32(
            false, a, false, b1, (short)0, acc1, false, false);
    }

    // Epilogue: D layout — VGPR i holds M = i (lanes 0-15) / i+8 (lanes 16-31),
    // N = lane&15 (+16 for the second tile).
    #pragma unroll
    for (int i = 0; i < 8; ++i) {
        const int me  = i + ((lane >= 16) ? 8 : 0);
        const int row = row0 + me;
        {   // tile 0: n in [0,16) -> alpha (0..9) / beta (10..15)
            const int n = n0;
            const float p = __expf(acc0[i] + bias[n]);
            ws_params[(size_t)row * 32 + n] = p;
        }
        {   // tile 1: n in [16,32) -> beta (16..19) / kappa (20..29)
            const int n = n0 + 16;
            if (n < N3K) {
                float p = __expf(acc1[i] + bias[n]);
                if (n >= 20) {
                    p += kappa_old[(size_t)row * K_SZ + (n - 20)];
                    out_kappa[(size_t)row * K_SZ + (n - 20)] = p;
                }
                ws_params[(size_t)row * 32 + n] = p;
            }
        }
    }
}

// ---------------------------------------------------------------------------
// Kernel 2: phi[b,l] = (L/len[b]) * sum_k alpha_k * exp(-beta_k*(kappa_k-l)^2)
// One block per batch row; 288 threads (9 waves) cover l=0..256.
// ---------------------------------------------------------------------------
__global__ __launch_bounds__(288) void window_phi(
    const float* __restrict__ ws_params, const float* __restrict__ text_lens,
    float* __restrict__ out_phi)
{
    __shared__ float ab[32];
    const int b = blockIdx.x;
    const int t = threadIdx.x;

    if (t < 32) ab[t] = ws_params[(size_t)b * 32 + t];
    __syncthreads();

    if (t <= L_SZ) {
        const float u = (float)t;
        float s = 0.0f;
        #pragma unroll
        for (int k = 0; k < K_SZ; ++k) {
            const float d = ab[20 + k] - u;
            s += ab[k] * __expf(-ab[10 + k] * d * d);
        }
        const float scale = (float)L_SZ / text_lens[b];
        out_phi[(size_t)b * (L_SZ + 1) + t] = s * scale;
    }
}

// ---------------------------------------------------------------------------
// Kernel 3: w[b,v] = sum_l phi[b,l] * onehots[b,l,v].  Bandwidth kernel:
// 335 MB streamed -> ~14.4 us roofline at 23.3 TB/s, dominates everything.
// One block per batch row; the (256x80) tile is read as 5120 float4s ->
// each of 256 threads issues 20 global_load_b128 at stride 256 (fully
// coalesced 512B per wave-request). idx -> l = idx/20, v = 4*(idx%20);
// v cycles with period 5 across the 20 steps, so 5 float4 accumulators,
// then 20 ds_add_f32 LDS atomics at the end.
// ---------------------------------------------------------------------------
__global__ __launch_bounds__(256) void window_w(
    const float* __restrict__ onehots, const float* __restrict__ phi,
    float* __restrict__ out_w)
{
    __shared__ float phi_s[L_SZ];
    __shared__ float w_s[V_SZ];
    const int b = blockIdx.x;
    const int t = threadIdx.x;

    phi_s[t] = phi[(size_t)b * (L_SZ + 1) + t];
    if (t < V_SZ) w_s[t] = 0.0f;
    __syncthreads();

    const float4* base = (const float4*)(onehots + (size_t)b * L_SZ * V_SZ);

    float4 acc[5];
    #pragma unroll
    for (int q = 0; q < 5; ++q) acc[q] = make_float4(0.f, 0.f, 0.f, 0.f);

    #pragma unroll
    for (int i = 0; i < 20; ++i) {
        const int idx = t + (i << 8);          // < 5120
        const float4 d = base[idx];
        const int l = idx / 20;                // row 0..255
        const float p = phi_s[l];
        const int q = i % 5;                   // same v-quad for i, i+5, ...
        acc[q].x += p * d.x;
        acc[q].y += p * d.y;
        acc[q].z += p * d.z;
        acc[q].w += p * d.w;
    }

    #pragma unroll
    for (int q = 0; q < 5; ++q) {
        const int v = ((t + (q << 8)) % 20) << 2;   // v-quad base for group q
        atomicAdd(&w_s[v + 0], acc[q].x);
        atomicAdd(&w_s[v + 1], acc[q].y);
        atomicAdd(&w_s[v + 2], acc[q].z);
        atomicAdd(&w_s[v + 3], acc[q].w);
    }
    __syncthreads();

    if (t < V_SZ) out_w[(size_t)b * V_SZ + t] = w_s[t];
}

// ---------------------------------------------------------------------------
extern "C" void kernel_launch(void* const* d_in, const int* in_sizes, int n_in,
                              void* d_out, int out_size, void* d_ws, size_t ws_size,
                              hipStream_t stream) {
    const float* x         = (const float*)d_in[0];  // (B, CELL)
    const float* kappa_old = (const float*)d_in[1];  // (B, K)
    const float* onehots   = (const float*)d_in[2];  // (B, L, V)
    const float* text_lens = (const float*)d_in[3];  // (B, 1)
    const float* W         = (const float*)d_in[4];  // (CELL, 3K)
    const float* bias      = (const float*)d_in[5];  // (3K,)

    float* out        = (float*)d_out;
    float* out_w      = out;                                   // B*V
    float* out_kappa  = out + (size_t)B_SZ * V_SZ;             // B*K
    float* out_phi    = out_kappa + (size_t)B_SZ * K_SZ;       // B*(L+1)

    float* ws_params  = (float*)d_ws;                          // B*32 floats

    // 1) fused GEMM + exp + kappa  (WMMA f32, pair-interleaved LDS W)
    window_gemm_wmma<<<B_SZ / (16 * 8), 256, 0, stream>>>(
        x, W, bias, kappa_old, ws_params, out_kappa);

    // 2) phi
    window_phi<<<B_SZ, 288, 0, stream>>>(ws_params, text_lens, out_phi);

    // 3) w (bandwidth-bound streaming reduction, b128 loads)
    window_w<<<B_SZ, 256, 0, stream>>>(onehots, out_phi, out_w);
}